// GCNBaseline_13469017440610
// MI455X (gfx1250) — compile-verified
//
#include <hip/hip_runtime.h>

#define N_NODES  50000
#define N_EDGES  800000
#define DIM      128
#define N_LAYERS 4
#define N_GRAPHS 512
#define N_COLS   4
#define VOCAB    128

typedef __attribute__((ext_vector_type(2))) float v2f;
typedef __attribute__((ext_vector_type(4))) float v4f;
typedef __attribute__((ext_vector_type(8))) float v8f;

// ---------------- embedding: h[n,d] = sum_c emb[c, x[n,c], d] ----------------
__global__ void embed_kernel(const int* __restrict__ x, const float* __restrict__ emb,
                             float* __restrict__ h) {
  int t = blockIdx.x * blockDim.x + threadIdx.x;
  if (t >= N_NODES * DIM) return;
  int n = t / DIM, d = t % DIM;
  float s = 0.f;
#pragma unroll
  for (int c = 0; c < N_COLS; ++c) {
    int v = x[n * N_COLS + c];
    s += emb[((size_t)c * VOCAB + v) * DIM + d];
  }
  h[t] = s;
}

// ---------------- degree / norm ----------------
__global__ void deg_init_kernel(float* __restrict__ deg) {
  int n = blockIdx.x * blockDim.x + threadIdx.x;
  if (n < N_NODES) deg[n] = 1.0f;  // self loop
}

__global__ void deg_edge_kernel(const int* __restrict__ ei, float* __restrict__ deg) {
  int e = blockIdx.x * blockDim.x + threadIdx.x;
  if (e < N_EDGES) unsafeAtomicAdd(&deg[ei[N_EDGES + e]], 1.0f);  // dst row
}

__global__ void dinv_kernel(float* __restrict__ deg) {
  int n = blockIdx.x * blockDim.x + threadIdx.x;
  if (n >= N_NODES) return;
  float d = deg[n];
  deg[n] = (d > 0.f) ? rsqrtf(d) : 0.f;  // in place: deg -> dinv
}

__global__ void norm_kernel(const int* __restrict__ ei, const float* __restrict__ dinv,
                            float* __restrict__ norm) {
  int e = blockIdx.x * blockDim.x + threadIdx.x;
  if (e >= N_EDGES) return;
  norm[e] = dinv[ei[e]] * dinv[ei[N_EDGES + e]];
}

// ---------------- WMMA fp32 GEMM: Cout[M,128] = A[M,128] @ B[128,128] (+bias) ----------------
// one wave (32 lanes) per 16x16 output tile; K loop in steps of 4 (V_WMMA_F32_16X16X4_F32)
__global__ void gemm_wmma_f32(const float* __restrict__ A, const float* __restrict__ B,
                              const float* __restrict__ bias, float* __restrict__ Cout) {
  int row0 = blockIdx.x * 16;
  int col0 = blockIdx.y * 16;
  int lane = threadIdx.x;
  int half = lane >> 4;   // 0: K pair {0,1}; 1: K pair {2,3}
  int lm   = lane & 15;

  const float* ap = A + (size_t)(row0 + lm) * DIM + half * 2;        // A rows, M=lm
  const float* bp = B + (size_t)(half * 2) * DIM + col0 + lm;        // B rows, N=lm

  v8f acc = {};
#pragma unroll
  for (int k = 0; k < DIM; k += 4) {
    v2f a = *(const v2f*)(ap + k);          // K=k+2*half, k+2*half+1
    v2f b;
    b.x = bp[k * DIM];                      // row k + 2*half
    b.y = bp[k * DIM + DIM];                // row k + 2*half + 1
    acc = __builtin_amdgcn_wmma_f32_16x16x4_f32(
        /*neg_a=*/false, a, /*neg_b=*/false, b,
        /*c_mod=*/(short)0, acc, /*reuse_a=*/false, /*reuse_b=*/false);
  }

  float badd = bias ? bias[col0 + lm] : 0.f;
#pragma unroll
  for (int r = 0; r < 8; ++r)
    Cout[(size_t)(row0 + half * 8 + r) * DIM + col0 + lm] = acc[r] + badd;
}

// ---------------- per-layer: init with bias + self-loop contribution ----------------
__global__ void init_self_bias_kernel(const float* __restrict__ xw, const float* __restrict__ dinv,
                                      const float* __restrict__ bl, float* __restrict__ nh) {
  int t = blockIdx.x * blockDim.x + threadIdx.x;
  if (t >= N_NODES * DIM) return;
  int n = t / DIM, d = t % DIM;
  float s = dinv[n];
  nh[t] = bl[d] + xw[t] * s * s;
}

// ---------------- edge scatter: nh[dst] += xw[src] * norm[e] ----------------
__global__ void scatter_kernel(const int* __restrict__ ei, const float* __restrict__ norm,
                               const float* __restrict__ xw, float* __restrict__ nh) {
  int t = blockIdx.x * blockDim.x + threadIdx.x;
  if (t >= N_EDGES * 32) return;
  int e  = t >> 5;
  int dg = (t & 31) << 2;  // 4 floats per lane
  int s = ei[e];
  int d = ei[N_EDGES + e];
  float w = norm[e];
  v4f v = *(const v4f*)(xw + (size_t)s * DIM + dg);
  float* dst = nh + (size_t)d * DIM + dg;
  unsafeAtomicAdd(dst + 0, v.x * w);
  unsafeAtomicAdd(dst + 1, v.y * w);
  unsafeAtomicAdd(dst + 2, v.z * w);
  unsafeAtomicAdd(dst + 3, v.w * w);
}

__global__ void relu_kernel(const float* __restrict__ nh, float* __restrict__ h) {
  int t = blockIdx.x * blockDim.x + threadIdx.x;
  if (t < N_NODES * DIM) h[t] = fmaxf(nh[t], 0.f);
}

// ---------------- mean pool ----------------
__global__ void pool_zero_kernel(float* __restrict__ pool, float* __restrict__ cnt) {
  int t = blockIdx.x * blockDim.x + threadIdx.x;
  if (t < N_GRAPHS * DIM) pool[t] = 0.f;
  else if (t < N_GRAPHS * DIM + N_GRAPHS) cnt[t - N_GRAPHS * DIM] = 0.f;
}

__global__ void pool_scatter_kernel(const int* __restrict__ batch, const float* __restrict__ h,
                                    float* __restrict__ pool) {
  int t = blockIdx.x * blockDim.x + threadIdx.x;
  if (t >= N_NODES * DIM) return;
  int n = t / DIM, d = t % DIM;
  unsafeAtomicAdd(&pool[(size_t)batch[n] * DIM + d], h[t]);
}

__global__ void cnt_scatter_kernel(const int* __restrict__ batch, float* __restrict__ cnt) {
  int n = blockIdx.x * blockDim.x + threadIdx.x;
  if (n < N_NODES) unsafeAtomicAdd(&cnt[batch[n]], 1.0f);
}

__global__ void pool_div_kernel(float* __restrict__ pool, const float* __restrict__ cnt) {
  int t = blockIdx.x * blockDim.x + threadIdx.x;
  if (t >= N_GRAPHS * DIM) return;
  pool[t] /= fmaxf(cnt[t / DIM], 1.0f);
}

// ---------------- host side ----------------
extern "C" void kernel_launch(void* const* d_in, const int* in_sizes, int n_in,
                              void* d_out, int out_size, void* d_ws, size_t ws_size,
                              hipStream_t stream) {
  const int*   x    = (const int*)d_in[0];
  const int*   ei   = (const int*)d_in[1];   // [2,E]: src at 0, dst at E
  const int*   batch= (const int*)d_in[2];
  const float* emb  = (const float*)d_in[3];
  const float* W    = (const float*)d_in[4]; // [L,128,128]
  const float* b    = (const float*)d_in[5]; // [L,128]
  const float* Wr   = (const float*)d_in[6]; // [128,128]
  const float* br   = (const float*)d_in[7]; // [128]
  float* out = (float*)d_out;

  char* ws = (char*)d_ws;
  const size_t NH = (size_t)N_NODES * DIM * sizeof(float);   // 25.6 MB
  float* h    = (float*)(ws);
  float* xw   = (float*)(ws + NH);
  float* nh   = (float*)(ws + 2 * NH);
  float* dinv = (float*)(ws + 3 * NH);                       // 200 KB (deg -> dinv)
  float* norm = (float*)(ws + 3 * NH + 256 * 1024);          // 3.2 MB
  float* pool = (float*)(ws + 3 * NH + 4 * 1024 * 1024);     // 256 KB
  float* cnt  = (float*)(ws + 3 * NH + 4 * 1024 * 1024 + 512 * 1024);

  const int B256 = 256;
  const int nd_blocks = (N_NODES * DIM + B256 - 1) / B256;
  const int n_blocks  = (N_NODES + B256 - 1) / B256;
  const int e_blocks  = (N_EDGES + B256 - 1) / B256;

  // 1. embedding
  embed_kernel<<<nd_blocks, B256, 0, stream>>>(x, emb, h);

  // 2. degrees + norms
  deg_init_kernel<<<n_blocks, B256, 0, stream>>>(dinv);
  deg_edge_kernel<<<e_blocks, B256, 0, stream>>>(ei, dinv);
  dinv_kernel<<<n_blocks, B256, 0, stream>>>(dinv);
  norm_kernel<<<e_blocks, B256, 0, stream>>>(ei, dinv, norm);

  // 3. GCN layers
  dim3 ggrid(N_NODES / 16, DIM / 16);  // 3125 x 8 tiles, one wave each
  const int sc_blocks = (N_EDGES * 32 + B256 - 1) / B256;
  for (int l = 0; l < N_LAYERS; ++l) {
    gemm_wmma_f32<<<ggrid, 32, 0, stream>>>(h, W + (size_t)l * DIM * DIM, nullptr, xw);
    init_self_bias_kernel<<<nd_blocks, B256, 0, stream>>>(xw, dinv, b + (size_t)l * DIM, nh);
    scatter_kernel<<<sc_blocks, B256, 0, stream>>>(ei, norm, xw, nh);
    relu_kernel<<<nd_blocks, B256, 0, stream>>>(nh, h);
  }

  // 4. mean pool
  pool_zero_kernel<<<(N_GRAPHS * DIM + N_GRAPHS + B256 - 1) / B256, B256, 0, stream>>>(pool, cnt);
  pool_scatter_kernel<<<nd_blocks, B256, 0, stream>>>(batch, h, pool);
  cnt_scatter_kernel<<<n_blocks, B256, 0, stream>>>(batch, cnt);
  pool_div_kernel<<<(N_GRAPHS * DIM + B256 - 1) / B256, B256, 0, stream>>>(pool, cnt);

  // 5. readout: out = pool @ Wr + br
  dim3 rgrid(N_GRAPHS / 16, DIM / 16);  // 32 x 8
  gemm_wmma_f32<<<rgrid, 32, 0, stream>>>(pool, Wr, br, out);
}